// OracleCT_Pillar_UnaryAttnPool_29618094474018
// MI455X (gfx1250) — compile-verified
//
#include <hip/hip_runtime.h>

typedef float v2f __attribute__((ext_vector_type(2)));
typedef float v8f __attribute__((ext_vector_type(8)));

#define C_DIM 1152
#define N_DIM 262144   // 64^3
#define D_DIM 30
#define ESTR  260      // padded LDS stride (floats) for e-tile -> conflict-free ds_load_b64
#define CTPW  9        // c-tiles (of 16) per wave: 72 tiles / 8 waves
#define CHUNKS 4       // n-chunks of 256 per block in the pooling kernel
#define PF_AHEAD 16    // prefetch distance in c-rows

__device__ __forceinline__ v8f vzero8() {
  v8f z = {0.f,0.f,0.f,0.f,0.f,0.f,0.f,0.f};
  return z;
}

// order-preserving float<->uint encoding for atomicMax on floats
__device__ __forceinline__ unsigned f32_ord(float f) {
  unsigned u = __float_as_uint(f);
  return (u & 0x80000000u) ? ~u : (u | 0x80000000u);
}
__device__ __forceinline__ float ord_f32(unsigned u) {
  return (u & 0x80000000u) ? __uint_as_float(u & 0x7FFFFFFFu)
                           : __uint_as_float(~u);
}

// Compute 32x32 tile of raw logits dot[d, n] for this wave's 32 n-columns.
// acc[m][j]: m = d-tile (d = m*16 + row), j = n-tile (n = n_wave + j*16 + col).
// A = swpad [16d x 4c] (pre-padded, rows 30/31 are zero -> no guards),
// B = feat [4c x 16n] (coalesced 64B half-wave rows).
__device__ __forceinline__ void dot_phase(const float* __restrict__ feat,
                                          const float* __restrict__ swp,
                                          int n_wave, int lane, v8f acc[2][2]) {
  const int ll = lane & 15;
  const int lh = lane >> 4;                 // 0: K={0,1}, 1: K={2,3}
  const long aOff0 = (long)ll * C_DIM + lh * 2;          // d = ll
  const long aOff1 = (long)(16 + ll) * C_DIM + lh * 2;   // d = 16 + ll
  const long b0n = (long)(lh * 2) * N_DIM + n_wave + ll; // row c+lh*2, col n
  const long b1n = b0n + 16;
  int c = 0;
  // main loop: unconditional prefetch PF_AHEAD rows ahead (always in-bounds)
  #pragma unroll 4
  for (; c < C_DIM - PF_AHEAD; c += 4) {
    v2f a0 = *(const v2f*)(swp + aOff0 + c);
    v2f a1 = *(const v2f*)(swp + aOff1 + c);
    const long bo = (long)c * N_DIM;
    v2f b0, b1;
    b0.x = feat[b0n + bo];
    b0.y = feat[b0n + bo + N_DIM];
    b1.x = feat[b1n + bo];
    b1.y = feat[b1n + bo + N_DIM];
    __builtin_prefetch(feat + b0n + bo + (long)PF_AHEAD * N_DIM, 0, 1);
    acc[0][0] = __builtin_amdgcn_wmma_f32_16x16x4_f32(false, a0, false, b0, (short)0, acc[0][0], false, false);
    acc[0][1] = __builtin_amdgcn_wmma_f32_16x16x4_f32(false, a0, false, b1, (short)0, acc[0][1], false, false);
    acc[1][0] = __builtin_amdgcn_wmma_f32_16x16x4_f32(false, a1, false, b0, (short)0, acc[1][0], false, false);
    acc[1][1] = __builtin_amdgcn_wmma_f32_16x16x4_f32(false, a1, false, b1, (short)0, acc[1][1], false, false);
  }
  // tail: last PF_AHEAD rows, no prefetch
  #pragma unroll
  for (; c < C_DIM; c += 4) {
    v2f a0 = *(const v2f*)(swp + aOff0 + c);
    v2f a1 = *(const v2f*)(swp + aOff1 + c);
    const long bo = (long)c * N_DIM;
    v2f b0, b1;
    b0.x = feat[b0n + bo];
    b0.y = feat[b0n + bo + N_DIM];
    b1.x = feat[b1n + bo];
    b1.y = feat[b1n + bo + N_DIM];
    acc[0][0] = __builtin_amdgcn_wmma_f32_16x16x4_f32(false, a0, false, b0, (short)0, acc[0][0], false, false);
    acc[0][1] = __builtin_amdgcn_wmma_f32_16x16x4_f32(false, a0, false, b1, (short)0, acc[0][1], false, false);
    acc[1][0] = __builtin_amdgcn_wmma_f32_16x16x4_f32(false, a1, false, b0, (short)0, acc[1][0], false, false);
    acc[1][1] = __builtin_amdgcn_wmma_f32_16x16x4_f32(false, a1, false, b1, (short)0, acc[1][1], false, false);
  }
}

// init: counters + pooled accumulator + zero-padded score_w copy [32 x 1152]
__global__ void k_init(const float* __restrict__ sw, unsigned* __restrict__ maxe,
                       float* __restrict__ sums, float* __restrict__ pool,
                       float* __restrict__ swpad) {
  int gi = blockIdx.x * blockDim.x + threadIdx.x;
  if (gi < 32) { maxe[gi] = 0x007FFFFFu; /* encode(-inf) */ sums[gi] = 0.f; }
  if (gi < 32 * C_DIM) {
    pool[gi] = 0.f;
    swpad[gi] = (gi < D_DIM * C_DIM) ? sw[gi] : 0.f;
  }
}

// Pass 1: per-disease global max of raw logits.
__global__ __launch_bounds__(256) void k_max(const float* __restrict__ feat,
                                             const float* __restrict__ swp,
                                             unsigned* __restrict__ maxe) {
  __shared__ float smax[8][32];
  const int tid = threadIdx.x;
  const int w = tid >> 5, lane = tid & 31;
  const int ll = lane & 15, lh = lane >> 4;
  const int n_wave = blockIdx.x * 256 + w * 32;
  v8f acc[2][2];
  #pragma unroll
  for (int m = 0; m < 2; ++m)
    #pragma unroll
    for (int j = 0; j < 2; ++j) acc[m][j] = vzero8();
  dot_phase(feat, swp, n_wave, lane, acc);
  // D layout: VGPR i, lanes0-15 -> d=m*16+i (n=lane); lanes16-31 -> d=m*16+i+8
  #pragma unroll
  for (int m = 0; m < 2; ++m) {
    #pragma unroll
    for (int i = 0; i < 8; ++i) {
      float v = fmaxf(acc[m][0][i], acc[m][1][i]);
      v = fmaxf(v, __shfl_xor(v, 1));
      v = fmaxf(v, __shfl_xor(v, 2));
      v = fmaxf(v, __shfl_xor(v, 4));
      v = fmaxf(v, __shfl_xor(v, 8));   // reduce over n within each lane-half
      if (ll == 0) smax[w][m * 16 + i + lh * 8] = v;
    }
  }
  __syncthreads();
  if (tid < 32) {
    float m = smax[0][tid];
    #pragma unroll
    for (int ww = 1; ww < 8; ++ww) m = fmaxf(m, smax[ww][tid]);
    atomicMax(&maxe[tid], f32_ord(m));
  }
}

// Pass 2: recompute logits, e = exp(clip((dot-max)/tau, -60, 0)),
// accumulate sum_e[d] and pooledT[d][c] = sum_n feat[c,n]*e[d,n] via WMMA.
__global__ __launch_bounds__(256) void k_pool(const float* __restrict__ feat,
                                              const float* __restrict__ swp,
                                              const float* __restrict__ tlogit,
                                              const unsigned* __restrict__ maxe,
                                              float* __restrict__ sums,
                                              float* __restrict__ pool) {
  __shared__ float dsmax[32];
  __shared__ float dsinv[32];
  __shared__ float elds[32 * ESTR];
  const int tid = threadIdx.x;
  const int w = tid >> 5, lane = tid & 31;
  const int ll = lane & 15, lh = lane >> 4;
  if (tid < 32) {
    if (tid < D_DIM) {
      float tau = 0.2f + 1.8f / (1.f + __expf(-tlogit[tid]));
      dsinv[tid] = 1.0f / tau;
      dsmax[tid] = ord_f32(maxe[tid]);
    } else {
      dsinv[tid] = 1.f;
      dsmax[tid] = 0.f;   // junk rows 30,31: bounded e<=1, never read out
    }
  }
  __syncthreads();

  v8f acc2[CTPW][2];
  #pragma unroll
  for (int t = 0; t < CTPW; ++t) { acc2[t][0] = vzero8(); acc2[t][1] = vzero8(); }
  float sacc[16];
  #pragma unroll
  for (int i = 0; i < 16; ++i) sacc[i] = 0.f;

  for (int chunk = 0; chunk < CHUNKS; ++chunk) {
    const int n0 = (blockIdx.x * CHUNKS + chunk) * 256;
    v8f acc[2][2];
    #pragma unroll
    for (int m = 0; m < 2; ++m) { acc[m][0] = vzero8(); acc[m][1] = vzero8(); }
    dot_phase(feat, swp, n0 + w * 32, lane, acc);

    // exponentiate, accumulate per-d partial sums, stage e into LDS [d][n_local]
    #pragma unroll
    for (int m = 0; m < 2; ++m) {
      #pragma unroll
      for (int i = 0; i < 8; ++i) {
        const int d = m * 16 + i + lh * 8;
        const float mx = dsmax[d], iv = dsinv[d];
        #pragma unroll
        for (int j = 0; j < 2; ++j) {
          float x = (acc[m][j][i] - mx) * iv;
          x = fminf(0.f, fmaxf(-60.f, x));
          float e = __expf(x);
          sacc[m * 8 + i] += e;
          elds[d * ESTR + w * 32 + j * 16 + ll] = e;
        }
      }
    }
    __syncthreads();

    // pooledT accumulation: D[16c x 16d] += A(feat 16c x 4n) * B(e 4n x 16d)
    #pragma unroll
    for (int t = 0; t < CTPW; ++t) {
      const int cb = (w * CTPW + t) * 16;
      const long arow = (long)(cb + ll) * N_DIM + n0 + lh * 2;
      for (int k = 0; k < 256; k += 4) {
        v2f a2 = *(const v2f*)(feat + arow + k);
        #pragma unroll
        for (int dt = 0; dt < 2; ++dt) {
          v2f b2 = *(const v2f*)&elds[(dt * 16 + ll) * ESTR + k + lh * 2];
          acc2[t][dt] = __builtin_amdgcn_wmma_f32_16x16x4_f32(
              false, a2, false, b2, (short)0, acc2[t][dt], false, false);
        }
      }
    }
    __syncthreads();   // before next chunk overwrites elds
  }

  // per-disease sum of exp
  #pragma unroll
  for (int m = 0; m < 2; ++m) {
    #pragma unroll
    for (int i = 0; i < 8; ++i) {
      float v = sacc[m * 8 + i];
      v += __shfl_xor(v, 1);
      v += __shfl_xor(v, 2);
      v += __shfl_xor(v, 4);
      v += __shfl_xor(v, 8);
      if (ll == 0) atomicAdd(&sums[m * 16 + i + lh * 8], v);
    }
  }
  // pooled partials: element (c,d) from D layout of acc2
  #pragma unroll
  for (int t = 0; t < CTPW; ++t) {
    #pragma unroll
    for (int dt = 0; dt < 2; ++dt) {
      #pragma unroll
      for (int i = 0; i < 8; ++i) {
        const int c = (w * CTPW + t) * 16 + i + lh * 8;
        const int d = dt * 16 + ll;
        atomicAdd(&pool[d * C_DIM + c], acc2[t][dt][i]);
      }
    }
  }
}

// Final per-disease head: out[d] = nan0( sum_c nan0(P[d,c]/S)*hw[d,c] + hb[d] )
__global__ __launch_bounds__(256) void k_out(const float* __restrict__ pool,
                                             const float* __restrict__ sums,
                                             const float* __restrict__ hw,
                                             const float* __restrict__ hb,
                                             float* __restrict__ out) {
  const int d = blockIdx.x;
  const int tid = threadIdx.x;
  __shared__ float red[8];
  const float S = fmaxf(sums[d], 1e-12f);
  const float inv = 1.0f / S;
  float p = 0.f;
  for (int c = tid; c < C_DIM; c += 256) {
    float z = pool[d * C_DIM + c] * inv;
    z = (z != z) ? 0.f : z;
    p += z * hw[d * C_DIM + c];
  }
  p += __shfl_xor(p, 1);
  p += __shfl_xor(p, 2);
  p += __shfl_xor(p, 4);
  p += __shfl_xor(p, 8);
  p += __shfl_xor(p, 16);
  if ((tid & 31) == 0) red[tid >> 5] = p;
  __syncthreads();
  if (tid == 0) {
    float r = 0.f;
    #pragma unroll
    for (int ww = 0; ww < 8; ++ww) r += red[ww];
    r += hb[d];
    out[d] = (r != r) ? 0.f : r;
  }
}

extern "C" void kernel_launch(void* const* d_in, const int* in_sizes, int n_in,
                              void* d_out, int out_size, void* d_ws, size_t ws_size,
                              hipStream_t stream) {
  const float* activ      = (const float*)d_in[0];
  const float* score_w    = (const float*)d_in[1];
  // d_in[2] = score_b: cancels in the softmax (shift-invariant per disease)
  const float* head_w     = (const float*)d_in[3];
  const float* head_b     = (const float*)d_in[4];
  const float* temp_logit = (const float*)d_in[5];
  float* out = (float*)d_out;

  char* ws = (char*)d_ws;
  unsigned* maxe  = (unsigned*)ws;                    // 32 u32
  float*    sums  = (float*)(ws + 128);               // 32 f32
  float*    pool  = (float*)(ws + 256);               // 32*1152 f32
  float*    swpad = (float*)(ws + 256 + 32 * C_DIM * 4); // 32*1152 f32, zero-padded

  k_init<<<144, 256, 0, stream>>>(score_w, maxe, sums, pool, swpad);
  k_max<<<N_DIM / 256, 256, 0, stream>>>(activ, swpad, maxe);
  k_pool<<<N_DIM / (256 * CHUNKS), 256, 0, stream>>>(activ, swpad, temp_logit,
                                                     maxe, sums, pool);
  k_out<<<D_DIM, 256, 0, stream>>>(pool, sums, head_w, head_b, out);
}